// PointPatchEmbed_32976758898981
// MI455X (gfx1250) — compile-verified
//
#include <hip/hip_runtime.h>
#include <cstdint>

typedef __attribute__((ext_vector_type(2))) float v2f;
typedef __attribute__((ext_vector_type(8))) float v8f;

#define NPTS   16384
#define NB     64
#define NS     64      // patches per batch
#define NK     32      // points per patch
#define NROWS  131072  // NB*NS*NK
#define EMB    384

// ---------------------------------------------------------------------------
// 1) Farthest point sampling: one block per batch, dist[] lives in LDS.
//    Matches jax scan: record current farthest, update min-dist, argmax.
// ---------------------------------------------------------------------------
__global__ void fps_kernel(const float* __restrict__ x,
                           int* __restrict__ fps_idx,
                           float* __restrict__ centers) {
    __shared__ float dist[NPTS];         // 64 KB (gfx1250 WGP has 320 KB LDS)
    __shared__ float red_v[32];
    __shared__ int   red_i[32];
    __shared__ float cen[3];
    __shared__ int   cur;

    const int b = blockIdx.x;
    const int t = threadIdx.x;
    const float* xb = x + (size_t)b * NPTS * 3;

    for (int i = t; i < NPTS; i += blockDim.x) dist[i] = 1e10f;
    if (t == 0) cur = 0;
    __syncthreads();

    for (int s = 0; s < NS; ++s) {
        if (t == 0) {
            const int f = cur;
            fps_idx[b * NS + s] = f;
            const float cx = xb[f * 3 + 0];
            const float cy = xb[f * 3 + 1];
            const float cz = xb[f * 3 + 2];
            cen[0] = cx; cen[1] = cy; cen[2] = cz;
            centers[(size_t)(b * NS + s) * 3 + 0] = cx;
            centers[(size_t)(b * NS + s) * 3 + 1] = cy;
            centers[(size_t)(b * NS + s) * 3 + 2] = cz;
        }
        __syncthreads();
        const float cx = cen[0], cy = cen[1], cz = cen[2];

        float bv = -1.0f; int bi = 0;
        for (int i = t; i < NPTS; i += blockDim.x) {
            const float dx = xb[i * 3 + 0] - cx;
            const float dy = xb[i * 3 + 1] - cy;
            const float dz = xb[i * 3 + 2] - cz;
            float d = dx * dx + dy * dy + dz * dz;
            d = fminf(dist[i], d);
            dist[i] = d;
            if (d > bv) { bv = d; bi = i; }   // ascending i -> first max kept
        }
        // wave32 argmax reduce (tie -> smaller index, matches jnp.argmax)
        for (int off = 16; off > 0; off >>= 1) {
            const float ov = __shfl_down(bv, off);
            const int   oi = __shfl_down(bi, off);
            if (ov > bv || (ov == bv && oi < bi)) { bv = ov; bi = oi; }
        }
        const int wid = t >> 5;
        if ((t & 31) == 0) { red_v[wid] = bv; red_i[wid] = bi; }
        __syncthreads();
        if (t == 0) {
            const int nw = blockDim.x >> 5;
            for (int w = 1; w < nw; ++w)
                if (red_v[w] > bv || (red_v[w] == bv && red_i[w] < bi)) {
                    bv = red_v[w]; bi = red_i[w];
                }
            cur = bi;
        }
        __syncthreads();
    }
}

// ---------------------------------------------------------------------------
// 2) K nearest neighbors per center: one block per (b,s); dist[] in LDS,
//    32 argmin sweeps with exclusion.
// ---------------------------------------------------------------------------
__global__ void topk_kernel(const float* __restrict__ x,
                            const float* __restrict__ centers,
                            int* __restrict__ knn_idx) {
    __shared__ float dist[NPTS];         // 64 KB
    __shared__ float red_v[8];
    __shared__ int   red_i[8];

    const int bs = blockIdx.x;           // b*NS + s
    const int b  = bs >> 6;
    const int t  = threadIdx.x;
    const float* xb = x + (size_t)b * NPTS * 3;
    const float cx = centers[(size_t)bs * 3 + 0];
    const float cy = centers[(size_t)bs * 3 + 1];
    const float cz = centers[(size_t)bs * 3 + 2];

    for (int i = t; i < NPTS; i += 256) {
        const float dx = xb[i * 3 + 0] - cx;
        const float dy = xb[i * 3 + 1] - cy;
        const float dz = xb[i * 3 + 2] - cz;
        dist[i] = dx * dx + dy * dy + dz * dz;
    }
    __syncthreads();

    for (int k = 0; k < NK; ++k) {
        float bv = 3.0e38f; int bi = 0;
        for (int i = t; i < NPTS; i += 256) {
            const float d = dist[i];
            if (d < bv) { bv = d; bi = i; }  // ascending i -> first min kept
        }
        for (int off = 16; off > 0; off >>= 1) {
            const float ov = __shfl_down(bv, off);
            const int   oi = __shfl_down(bi, off);
            if (ov < bv || (ov == bv && oi < bi)) { bv = ov; bi = oi; }
        }
        if ((t & 31) == 0) { red_v[t >> 5] = bv; red_i[t >> 5] = bi; }
        __syncthreads();
        if (t == 0) {
            for (int w = 1; w < 8; ++w)
                if (red_v[w] < bv || (red_v[w] == bv && red_i[w] < bi)) {
                    bv = red_v[w]; bi = red_i[w];
                }
            knn_idx[(size_t)bs * NK + k] = bi;
            dist[bi] = 3.0e38f;              // exclude selected point
        }
        __syncthreads();
    }
}

// ---------------------------------------------------------------------------
// 3) Gather + recenter + (3 -> 64) matmul + bias. K=3 is too small for WMMA;
//    pure VALU with W1/b1 cached in LDS.
// ---------------------------------------------------------------------------
__global__ void stage1_kernel(const float* __restrict__ x,
                              const float* __restrict__ centers,
                              const int* __restrict__ knn_idx,
                              const float* __restrict__ W1,
                              const float* __restrict__ b1,
                              float* __restrict__ h1) {
    __shared__ float w[64 * 3];
    __shared__ float bb[64];
    const int t = threadIdx.x;
    if (t < 192) w[t] = W1[t];
    if (t >= 192 && t < 256) bb[t - 192] = b1[t - 192];
    __syncthreads();

    const int row = blockIdx.x * 256 + t;        // 0 .. NROWS-1
    const int bs  = row >> 5;
    const int b   = bs >> 6;
    const int p   = knn_idx[row];
    const float* xp = x + ((size_t)b * NPTS + p) * 3;
    const float dx = xp[0] - centers[(size_t)bs * 3 + 0];
    const float dy = xp[1] - centers[(size_t)bs * 3 + 1];
    const float dz = xp[2] - centers[(size_t)bs * 3 + 2];

    float* out = h1 + (size_t)row * 64;
#pragma unroll 8
    for (int o = 0; o < 64; ++o)
        out[o] = fmaf(w[o * 3 + 0], dx,
                 fmaf(w[o * 3 + 1], dy,
                 fmaf(w[o * 3 + 2], dz, bb[o])));
}

// ---------------------------------------------------------------------------
// 4) BatchNorm statistics: one block per channel -> mean, var (E[x^2]-m^2).
// ---------------------------------------------------------------------------
template <int C>
__global__ void bn_stats_kernel(const float* __restrict__ h,
                                float* __restrict__ meanvar) {
    const int c = blockIdx.x;
    const int t = threadIdx.x;
    float s = 0.f, s2 = 0.f;
    for (int r = t; r < NROWS; r += 256) {
        const float v = h[(size_t)r * C + c];
        s += v; s2 += v * v;
    }
    for (int off = 16; off > 0; off >>= 1) {
        s  += __shfl_down(s, off);
        s2 += __shfl_down(s2, off);
    }
    __shared__ float as[8], as2[8];
    if ((t & 31) == 0) { as[t >> 5] = s; as2[t >> 5] = s2; }
    __syncthreads();
    if (t == 0) {
        for (int w = 1; w < 8; ++w) { s += as[w]; s2 += as2[w]; }
        const float inv = 1.0f / (float)NROWS;
        const float m = s * inv;
        meanvar[c]     = m;
        meanvar[C + c] = s2 * inv - m * m;
    }
}

template <int C>
__global__ void bn_finalize_kernel(const float* __restrict__ meanvar,
                                   const float* __restrict__ g,
                                   const float* __restrict__ be,
                                   float* __restrict__ scale,
                                   float* __restrict__ shift) {
    const int c = threadIdx.x;
    if (c < C) {
        const float m  = meanvar[c];
        const float v  = meanvar[C + c];
        const float sc = g[c] * rsqrtf(v + 1e-5f);
        scale[c] = sc;
        shift[c] = be[c] - m * sc;
    }
}

// ---------------------------------------------------------------------------
// 5) GEMM2: relu(bn(h1)) [131072x64] @ W2^T [64->128], fp32 WMMA 16x16x4.
//    One block (8 waves) per 16-row M tile: A tile (BN+ReLU applied) and the
//    whole W2 staged in LDS (padded stride 65 -> conflict-free row reads);
//    inner loop is pure ds_load_b64 + v_wmma. Wave w owns N tile w.
// ---------------------------------------------------------------------------
__global__ void gemm2_kernel(const float* __restrict__ h1,
                             const float* __restrict__ W2,
                             const float* __restrict__ b2,
                             const float* __restrict__ scale1,
                             const float* __restrict__ shift1,
                             float* __restrict__ h2) {
    __shared__ float As[16 * 65];        //  4.2 KB
    __shared__ float Ws[128 * 65];       // 33.3 KB
    const int t  = threadIdx.x;
    const int mt = blockIdx.x;           // 8192 M tiles

    // stage A with BN+ReLU fused (1024 elements)
    for (int i = t; i < 16 * 64; i += 256) {
        const int r = i >> 6, k = i & 63;
        As[r * 65 + k] =
            fmaxf(fmaf(h1[(size_t)(mt * 16 + r) * 64 + k], scale1[k], shift1[k]), 0.f);
    }
    // stage all of W2 (8192 elements)
    for (int i = t; i < 128 * 64; i += 256) {
        const int r = i >> 6, k = i & 63;
        Ws[r * 65 + k] = W2[i];
    }
    __syncthreads();

    const int lane = t & 31;
    const int nt   = t >> 5;             // wave id = N tile (0..7)
    const int half = lane >> 4, lid = lane & 15;
    const int n = nt * 16 + lid;

    v8f c;
    const float bias = b2[n];
#pragma unroll
    for (int r = 0; r < 8; ++r) c[r] = bias;

    const float* arow = As + lid * 65;
    const float* brow = Ws + n * 65;

#pragma unroll
    for (int k0 = 0; k0 < 64; k0 += 4) {
        const int kk = k0 + 2 * half;    // A/B layout: K = 2*(lane/16) + vgpr
        v2f a, bv;
        a.x  = arow[kk];
        a.y  = arow[kk + 1];
        bv.x = brow[kk];
        bv.y = brow[kk + 1];
        c = __builtin_amdgcn_wmma_f32_16x16x4_f32(false, a, false, bv,
                                                  (short)0, c, false, false);
    }
#pragma unroll
    for (int r = 0; r < 8; ++r) {
        const int m = mt * 16 + r + 8 * half;    // C/D layout: M = r + 8*half
        h2[(size_t)m * 128 + n] = c[r];
    }
}

// ---------------------------------------------------------------------------
// 6) GEMM3 fused with K-max-pool: relu(bn(h2)) [131072x128] @ W3^T [128->384].
//    One 768-thread block (24 waves) per patch: the patch's 32x128 activations
//    are BN+ReLU'd into LDS once and shared by all 24 N tiles; W3 rows stream
//    from global (L2-resident, each b64 load feeds two WMMAs). Per-wave max
//    over its 16 M values + cross-half shuffle max; h3 never touches HBM.
// ---------------------------------------------------------------------------
__global__ void gemm3_kernel(const float* __restrict__ h2,
                             const float* __restrict__ W3,
                             const float* __restrict__ b3,
                             const float* __restrict__ scale2,
                             const float* __restrict__ shift2,
                             float* __restrict__ emb) {
    __shared__ float As[32 * 132];       // 16.9 KB, padded stride (4lid+kk banks)
    const int t   = threadIdx.x;         // 0..767
    const int pid = blockIdx.x;          // 4096 patches

    // stage patch activations with BN+ReLU fused (4096 elements)
    for (int i = t; i < 32 * 128; i += 768) {
        const int r = i >> 7, k = i & 127;
        As[r * 132 + k] =
            fmaxf(fmaf(h2[(size_t)(pid * NK + r) * 128 + k], scale2[k], shift2[k]), 0.f);
    }
    __syncthreads();

    const int lane = t & 31;
    const int nt   = t >> 5;             // wave id = N tile (0..23)
    const int half = lane >> 4, lid = lane & 15;
    const int n = nt * 16 + lid;

    const float bias = b3[n];
    v8f c0, c1;
#pragma unroll
    for (int r = 0; r < 8; ++r) { c0[r] = bias; c1[r] = bias; }

    const float* a0  = As + lid * 132;          // rows 0..15 of the patch
    const float* a1  = As + (16 + lid) * 132;   // rows 16..31
    const float* brw = W3 + (size_t)n * 128;

#pragma unroll 8
    for (int k0 = 0; k0 < 128; k0 += 4) {
        const int kk = k0 + 2 * half;
        v2f bv;
        bv.x = brw[kk];
        bv.y = brw[kk + 1];
        v2f aa;
        aa.x = a0[kk];
        aa.y = a0[kk + 1];
        c0 = __builtin_amdgcn_wmma_f32_16x16x4_f32(false, aa, false, bv,
                                                   (short)0, c0, false, false);
        aa.x = a1[kk];
        aa.y = a1[kk + 1];
        c1 = __builtin_amdgcn_wmma_f32_16x16x4_f32(false, aa, false, bv,
                                                   (short)0, c1, false, false);
    }
    // max over this lane's 16 M values (all WMMAs done -> divergence is safe)
    float mx = c0[0];
#pragma unroll
    for (int r = 1; r < 8; ++r) mx = fmaxf(mx, c0[r]);
#pragma unroll
    for (int r = 0; r < 8; ++r) mx = fmaxf(mx, c1[r]);
    // combine the two lane halves (lanes l and l^16 hold disjoint M sets)
    mx = fmaxf(mx, __shfl_xor(mx, 16));
    if (half == 0)
        emb[(size_t)pid * EMB + n] = mx;
}

// ---------------------------------------------------------------------------
extern "C" void kernel_launch(void* const* d_in, const int* in_sizes, int n_in,
                              void* d_out, int out_size, void* d_ws, size_t ws_size,
                              hipStream_t stream) {
    const float* x   = (const float*)d_in[0];
    const float* W1  = (const float*)d_in[1];
    const float* b1  = (const float*)d_in[2];
    const float* g1  = (const float*)d_in[3];
    const float* be1 = (const float*)d_in[4];
    const float* W2  = (const float*)d_in[5];
    const float* b2  = (const float*)d_in[6];
    const float* g2  = (const float*)d_in[7];
    const float* be2 = (const float*)d_in[8];
    const float* W3  = (const float*)d_in[9];
    const float* b3  = (const float*)d_in[10];

    float* emb     = (float*)d_out;                       // (64,64,384)
    float* centers = emb + (size_t)NB * NS * EMB;         // (64,64,3)

    char*  ws      = (char*)d_ws;
    int*   fps_idx = (int*)ws;                                      // 16 KB
    int*   knn_idx = (int*)(ws + (16 << 10));                       // 512 KB
    float* h1      = (float*)(ws + (1 << 20));                      // 32 MB
    float* h2      = (float*)(ws + (1 << 20) + ((size_t)32 << 20)); // 64 MB
    float* mv1     = (float*)(ws + (1 << 20) + ((size_t)96 << 20));
    float* mv2     = mv1 + 2 * 64;
    float* sc1     = mv2 + 2 * 128;
    float* sh1     = sc1 + 64;
    float* sc2     = sh1 + 64;
    float* sh2     = sc2 + 128;

    fps_kernel<<<NB, 1024, 0, stream>>>(x, fps_idx, centers);
    topk_kernel<<<NB * NS, 256, 0, stream>>>(x, centers, knn_idx);
    stage1_kernel<<<NROWS / 256, 256, 0, stream>>>(x, centers, knn_idx, W1, b1, h1);
    bn_stats_kernel<64><<<64, 256, 0, stream>>>(h1, mv1);
    bn_finalize_kernel<64><<<1, 64, 0, stream>>>(mv1, g1, be1, sc1, sh1);
    gemm2_kernel<<<NROWS / 16, 256, 0, stream>>>(h1, W2, b2, sc1, sh1, h2);
    bn_stats_kernel<128><<<128, 256, 0, stream>>>(h2, mv2);
    bn_finalize_kernel<128><<<1, 128, 0, stream>>>(mv2, g2, be2, sc2, sh2);
    gemm3_kernel<<<NB * NS, 768, 0, stream>>>(h2, W3, b3, sc2, sh2, emb);
}